// AccessibilityGNNCorrector_66065186947677
// MI455X (gfx1250) — compile-verified
//
#include <hip/hip_runtime.h>
#include <math.h>

// AccessibilityGNNCorrector for gfx1250 (MI455X).
// GEMMs via v_wmma_f32_16x16x4_f32 (fp32 matrix pipe, exact fp32 numerics),
// 32 output rows per wave, B fragments pre-packed so each is one b64 load:
// inner loop = 10 b64 loads : 16 WMMAs (WMMA-issue bound, not VMEM bound).
// Edge aggregation: wave-per-edge float4 gather + global_atomic_add_f32 scatter
// (node feature matrices are 51.2MB -> fully L2-resident on 192MB L2).

#define HID 128
#define EPSV 1e-5f

typedef __attribute__((ext_vector_type(2))) float v2f;
typedef __attribute__((ext_vector_type(8))) float v8f;

// ---------------------------------------------------------------- utilities
__global__ void zero_f32(float* __restrict__ p, int n) {
    int i = blockIdx.x * blockDim.x + threadIdx.x;
    if (i < n) p[i] = 0.0f;
}

__global__ void deg_accum(const int* __restrict__ dst, float* __restrict__ deg, int E) {
    int e = blockIdx.x * blockDim.x + threadIdx.x;
    if (e < E) atomicAdd(&deg[dst[e]], 1.0f);
}

__global__ void dinv_finalize(float* __restrict__ dinv, int n) {
    int i = blockIdx.x * blockDim.x + threadIdx.x;
    if (i < n) dinv[i] = rsqrtf(dinv[i] + 1.0f);
}

// ------------------------------------------------- W -> WMMA-fragment order
// Wf layout: pair index p = (k>>2)*2 + ((k&3)>>1), lane col, then the two
// contiguous K elements. Lane (half,l16) of k-step s reads b64 at
// ((s*2+half)*FOUT + col)*2.
template <int K, int FOUT>
__global__ void pack_w(const float* __restrict__ W, float* __restrict__ Wf) {
    int i = blockIdx.x * blockDim.x + threadIdx.x;
    if (i < K * FOUT) {
        const int k = i / FOUT, col = i - k * FOUT;
        const int p = ((k >> 2) << 1) | ((k & 3) >> 1);
        const int sel = k & 1;
        Wf[(((size_t)p * FOUT + col) << 1) + sel] = W[i];
    }
}

// ---------------------------------------------------------------- WMMA GEMM
// C[rows x FOUT] = A[rows x K] @ W[K x FOUT]  (+ bias, relu if BIAS_RELU)
// One wave computes a 32 x FOUT strip (two 16-row tiles sharing B fragments).
// rows must be a multiple of 32.
template <int K, int FOUT, bool BIAS_RELU>
__global__ __launch_bounds__(128) void gemm_wmma(const float* __restrict__ A,
                                                 const float* __restrict__ Wf,
                                                 const float* __restrict__ bias,
                                                 float* __restrict__ C, int rows) {
    const int lane = threadIdx.x & 31;
    const int wave = threadIdx.x >> 5;
    const int half = lane >> 4;   // 0: lanes 0-15, 1: lanes 16-31
    const int l16  = lane & 15;
    const int m0 = (blockIdx.x * 4 + wave) * 32;
    if (m0 >= rows) return;       // whole-wave guard; EXEC all-ones for WMMA

    constexpr int NT = FOUT / 16;
    v8f acc0[NT] = {};            // rows m0 .. m0+15
    v8f acc1[NT] = {};            // rows m0+16 .. m0+31

    const float* arow0 = A + (size_t)(m0 + l16) * K + half * 2;
    const float* arow1 = arow0 + (size_t)16 * K;
    const float* wp = Wf + ((size_t)half * FOUT + l16) * 2;

    for (int k0 = 0; k0 < K; k0 += 4) {
        const float2 av0 = *(const float2*)(arow0 + k0);
        const float2 av1 = *(const float2*)(arow1 + k0);
        v2f a0 = {av0.x, av0.y};
        v2f a1 = {av1.x, av1.y};
#pragma unroll
        for (int t = 0; t < NT; ++t) {
            const float2 bv = *(const float2*)(wp + t * 32);
            v2f b = {bv.x, bv.y};
            acc0[t] = __builtin_amdgcn_wmma_f32_16x16x4_f32(
                false, a0, false, b, (short)0, acc0[t], false, false);
            acc1[t] = __builtin_amdgcn_wmma_f32_16x16x4_f32(
                false, a1, false, b, (short)0, acc1[t], false, false);
        }
        wp += 4 * FOUT;   // advance two K-pair rows
    }

#pragma unroll
    for (int t = 0; t < NT; ++t) {
        const int col = t * 16 + l16;
        float bv = 0.0f;
        if constexpr (BIAS_RELU) bv = bias[col];
#pragma unroll
        for (int g = 0; g < 8; ++g) {
            const int row = m0 + g + half * 8;
            float v0 = acc0[t][g];
            float v1 = acc1[t][g];
            if constexpr (BIAS_RELU) {
                v0 = fmaxf(v0 + bv, 0.0f);
                v1 = fmaxf(v1 + bv, 0.0f);
            }
            C[(size_t)row * FOUT + col] = v0;
            C[(size_t)(row + 16) * FOUT + col] = v1;
        }
    }
}

// ------------------------------------------------------- GCN aggregation
// agg[i,:] = hw[i,:] * dinv[i]^2   (self-loop term; also initializes agg)
template <int F>
__global__ void self_init(float* __restrict__ agg, const float* __restrict__ hw,
                          const float* __restrict__ dinv, int n) {
    size_t i = (size_t)blockIdx.x * blockDim.x + threadIdx.x;
    if (i < (size_t)n * F) {
        const int row = (int)(i / F);
        const float d = dinv[row];
        agg[i] = hw[i] * d * d;
    }
}

// agg[dst,:] += hw[src,:] * dinv[src]*dinv[dst]   (one wave per edge)
template <int F>
__global__ void edge_scatter(float* __restrict__ agg, const float* __restrict__ hw,
                             const int* __restrict__ src, const int* __restrict__ dst,
                             const float* __restrict__ dinv, int E) {
    const int lane = threadIdx.x & 31;
    const long e = ((long)blockIdx.x * blockDim.x + threadIdx.x) >> 5;
    if (e >= E) return;
    // e is wave-uniform: force s/d into SGPRs so gather/scatter use saddr form
    const int s = __builtin_amdgcn_readfirstlane(src[e]);
    const int d = __builtin_amdgcn_readfirstlane(dst[e]);
    const float c = dinv[s] * dinv[d];
    constexpr int V = F / 32;  // floats per lane (4 for F=128, 2 for F=64)
    const float* hp = hw + (size_t)s * F + lane * V;
    float* ap = agg + (size_t)d * F + lane * V;
    if constexpr (V == 4) {
        const float4 v = *(const float4*)hp;
        atomicAdd(ap + 0, v.x * c);
        atomicAdd(ap + 1, v.y * c);
        atomicAdd(ap + 2, v.z * c);
        atomicAdd(ap + 3, v.w * c);
    } else {
        const float2 v = *(const float2*)hp;
        atomicAdd(ap + 0, v.x * c);
        atomicAdd(ap + 1, v.y * c);
    }
}

// ------------------------------------------------------- BatchNorm (train-mode stats)
// stats[0..127] += column sums, stats[128..255] += column sums of squares
__global__ void col_stats(const float* __restrict__ agg, float* __restrict__ stats, int n) {
    const int gid = blockIdx.x * blockDim.x + threadIdx.x;
    const int col = gid & 127;
    const int group = gid >> 7;
    const int ngroups = (gridDim.x * blockDim.x) >> 7;
    float s = 0.0f, q = 0.0f;
    for (int r = group; r < n; r += ngroups) {
        const float v = agg[(size_t)r * HID + col];
        s += v;
        q = fmaf(v, v, q);
    }
    atomicAdd(&stats[col], s);
    atomicAdd(&stats[HID + col], q);
}

// Note: pre-BN bias (b1/b2) cancels under mean subtraction -> omitted entirely.
__global__ void bn_finalize(const float* __restrict__ stats, const float* __restrict__ gamma,
                            const float* __restrict__ beta, float* __restrict__ scl,
                            float* __restrict__ sh, int n) {
    const int c = threadIdx.x;
    if (c < HID) {
        const float inv_n = 1.0f / (float)n;
        const float mean = stats[c] * inv_n;
        const float var = stats[HID + c] * inv_n - mean * mean;
        const float s = rsqrtf(var + EPSV) * gamma[c];
        scl[c] = s;
        sh[c] = beta[c] - mean * s;
    }
}

// h = relu(h*scl + sh) [+ res]   (in place)
template <bool RES>
__global__ void bn_apply(float* __restrict__ h, const float* __restrict__ scl,
                         const float* __restrict__ sh, const float* __restrict__ res,
                         size_t total) {
    size_t i = (size_t)blockIdx.x * blockDim.x + threadIdx.x;
    if (i < total) {
        const int col = (int)(i & (HID - 1));
        float v = fmaxf(fmaf(h[i], scl[col], sh[col]), 0.0f);
        if constexpr (RES) v += res[i];
        h[i] = v;
    }
}

// out[i] = tanh( relu(agg3[i,:] + b3) . Wh + bh ) * scale
__global__ void head_kernel(const float* __restrict__ agg3, const float* __restrict__ b3,
                            const float* __restrict__ Wh, const float* __restrict__ bh,
                            const float* __restrict__ scale, float* __restrict__ out, int n) {
    const int i = blockIdx.x * blockDim.x + threadIdx.x;
    if (i >= n) return;
    const float* r = agg3 + (size_t)i * 64;
    float acc = 0.0f;
#pragma unroll 16
    for (int f = 0; f < 64; ++f) {
        const float v = fmaxf(r[f] + b3[f], 0.0f);
        acc = fmaf(v, Wh[f], acc);
    }
    out[i] = tanhf(acc + bh[0]) * scale[0];
}

// ---------------------------------------------------------------- launcher
extern "C" void kernel_launch(void* const* d_in, const int* in_sizes, int n_in,
                              void* d_out, int out_size, void* d_ws, size_t ws_size,
                              hipStream_t stream) {
    const float* x  = (const float*)d_in[0];
    const int*   ei = (const int*)d_in[1];
    const float* Wp = (const float*)d_in[2];
    const float* bp = (const float*)d_in[3];
    const float* W1 = (const float*)d_in[4];
    // b1 = d_in[5]: cancels under BN
    const float* g1 = (const float*)d_in[6];
    const float* be1 = (const float*)d_in[7];
    const float* W2 = (const float*)d_in[8];
    // b2 = d_in[9]: cancels under BN
    const float* g2 = (const float*)d_in[10];
    const float* be2 = (const float*)d_in[11];
    const float* W3 = (const float*)d_in[12];
    const float* b3 = (const float*)d_in[13];
    const float* Wh = (const float*)d_in[14];
    const float* bh = (const float*)d_in[15];
    const float* sc = (const float*)d_in[16];

    const int N = in_sizes[0] / 256;   // 100000 (multiple of 32)
    const int E = in_sizes[1] / 2;     // 3200000
    const int* src = ei;
    const int* dst = ei + E;

    char* ws = (char*)d_ws;
    const size_t BIG = (size_t)N * HID * sizeof(float);   // 51.2 MB
    float* bufA  = (float*)(ws);
    float* bufB  = (float*)(ws + BIG);
    float* bufC  = (float*)(ws + 2 * BIG);
    float* dinv  = (float*)(ws + 3 * BIG);
    float* stats = dinv + N;
    float* scl   = stats + 2 * HID;
    float* shf   = scl + HID;
    float* Wf    = shf + HID;          // packed weights, up to 256*128 floats

    auto cdiv = [](long a, long b) { return (unsigned)((a + b - 1) / b); };
    const unsigned gemm_blocks = cdiv(N, 128);                // 4 waves * 32 rows
    const unsigned elem128 = cdiv((long)N * HID, 256);
    const unsigned elem64  = cdiv((long)N * 64, 256);
    const unsigned edge_blk = cdiv((long)E * 32, 256);        // wave per edge

    // degrees -> dinv (in place)
    zero_f32<<<cdiv(N, 256), 256, 0, stream>>>(dinv, N);
    deg_accum<<<cdiv(E, 256), 256, 0, stream>>>(dst, dinv, E);
    dinv_finalize<<<cdiv(N, 256), 256, 0, stream>>>(dinv, N);

    // projection: h = relu(x @ Wp + bp)          -> bufA
    pack_w<256, 128><<<cdiv(256 * 128, 256), 256, 0, stream>>>(Wp, Wf);
    gemm_wmma<256, 128, true><<<gemm_blocks, 128, 0, stream>>>(x, Wf, bp, bufA, N);

    // ---- GCN layer 1 ----
    pack_w<128, 128><<<cdiv(128 * 128, 256), 256, 0, stream>>>(W1, Wf);
    gemm_wmma<128, 128, false><<<gemm_blocks, 128, 0, stream>>>(bufA, Wf, nullptr, bufB, N);
    self_init<128><<<elem128, 256, 0, stream>>>(bufC, bufB, dinv, N);
    edge_scatter<128><<<edge_blk, 256, 0, stream>>>(bufC, bufB, src, dst, dinv, E);
    zero_f32<<<1, 256, 0, stream>>>(stats, 2 * HID);
    col_stats<<<512, 256, 0, stream>>>(bufC, stats, N);
    bn_finalize<<<1, 128, 0, stream>>>(stats, g1, be1, scl, shf, N);
    bn_apply<false><<<elem128, 256, 0, stream>>>(bufC, scl, shf, nullptr, (size_t)N * HID);
    // h1 = bufC

    // ---- GCN layer 2 (+ residual) ----
    pack_w<128, 128><<<cdiv(128 * 128, 256), 256, 0, stream>>>(W2, Wf);
    gemm_wmma<128, 128, false><<<gemm_blocks, 128, 0, stream>>>(bufC, Wf, nullptr, bufA, N);
    self_init<128><<<elem128, 256, 0, stream>>>(bufB, bufA, dinv, N);
    edge_scatter<128><<<edge_blk, 256, 0, stream>>>(bufB, bufA, src, dst, dinv, E);
    zero_f32<<<1, 256, 0, stream>>>(stats, 2 * HID);
    col_stats<<<512, 256, 0, stream>>>(bufB, stats, N);
    bn_finalize<<<1, 128, 0, stream>>>(stats, g2, be2, scl, shf, N);
    bn_apply<true><<<elem128, 256, 0, stream>>>(bufB, scl, shf, bufC, (size_t)N * HID);
    // h2 = bufB

    // ---- GCN layer 3 (128 -> 64) ----
    pack_w<128, 64><<<cdiv(128 * 64, 256), 256, 0, stream>>>(W3, Wf);
    gemm_wmma<128, 64, false><<<gemm_blocks, 128, 0, stream>>>(bufB, Wf, nullptr, bufA, N);
    self_init<64><<<elem64, 256, 0, stream>>>(bufC, bufA, dinv, N);
    edge_scatter<64><<<edge_blk, 256, 0, stream>>>(bufC, bufA, src, dst, dinv, E);

    // head: out = tanh(relu(agg3 + b3) @ Wh + bh) * scale
    head_kernel<<<cdiv(N, 256), 256, 0, stream>>>(bufC, b3, Wh, bh, sc, (float*)d_out, N);
}